// InterpolationStage_6347961663491
// MI455X (gfx1250) — compile-verified
//
#include <hip/hip_runtime.h>

typedef __attribute__((ext_vector_type(2))) float v2f;
typedef __attribute__((ext_vector_type(8))) float v8f;

#define NC   4096     // coarse points
#define CC   512      // coarse feature dim
#define CFD  256      // fine feature dim
#define CIN  768      // CC + CFD
#define CO   256      // output dim
#define LN_EPS 1e-5f

// ---------------------------------------------------------------------------
// Kernel 1: brute-force 3-NN + inverse-squared-distance weights.
// One lane per fine point; coarse positions staged once into LDS (48 KB).
// ---------------------------------------------------------------------------
__global__ __launch_bounds__(256) void knn_kernel(
    const float* __restrict__ coarse_pos,   // [NC,3]
    const float* __restrict__ fine_pos,     // [Nf,3]
    int*   __restrict__ knn_idx,            // [Nf,3]
    float* __restrict__ knn_w,              // [Nf,3] (normalized)
    int Nf)
{
    __shared__ float sc[3][NC];
    const int tid = threadIdx.x;

    for (int t = tid; t < NC * 3; t += 256) {
        sc[t % 3][t / 3] = coarse_pos[t];   // coalesced global read
    }
    __syncthreads();

    const int n = blockIdx.x * 256 + tid;
    if (n >= Nf) return;

    const float fx = fine_pos[n * 3 + 0];
    const float fy = fine_pos[n * 3 + 1];
    const float fz = fine_pos[n * 3 + 2];

    float d0 = 3.4e38f, d1 = 3.4e38f, d2 = 3.4e38f;
    int   i0 = 0, i1 = 0, i2 = 0;

    #pragma unroll 4
    for (int c = 0; c < NC; ++c) {
        const float dx = fx - sc[0][c];     // LDS broadcast (all lanes same c)
        const float dy = fy - sc[1][c];
        const float dz = fz - sc[2][c];
        const float d  = dx * dx + dy * dy + dz * dz;
        if (d < d2) {                       // strict '<' keeps earliest index on ties
            d2 = d; i2 = c;
            if (d2 < d1) {
                float td = d1; d1 = d2; d2 = td;
                int   ti = i1; i1 = i2; i2 = ti;
                if (d1 < d0) {
                    td = d0; d0 = d1; d1 = td;
                    ti = i0; i0 = i1; i1 = ti;
                }
            }
        }
    }

    const float w0 = 1.0f / fmaxf(d0, 1e-16f);
    const float w1 = 1.0f / fmaxf(d1, 1e-16f);
    const float w2 = 1.0f / fmaxf(d2, 1e-16f);
    const float inv = 1.0f / (w0 + w1 + w2);

    knn_idx[n * 3 + 0] = i0;
    knn_idx[n * 3 + 1] = i1;
    knn_idx[n * 3 + 2] = i2;
    knn_w[n * 3 + 0] = w0 * inv;
    knn_w[n * 3 + 1] = w1 * inv;
    knn_w[n * 3 + 2] = w2 * inv;
}

// ---------------------------------------------------------------------------
// Kernel 2: fused [gather+blend | concat] -> f32 WMMA GEMM -> bias/ReLU -> LN.
// Block = 256 threads (8 wave32), one 16-row M-tile per block.
// Wave w owns output columns [32w, 32w+32): two 16x16 WMMA N-tiles.
// ---------------------------------------------------------------------------
__global__ __launch_bounds__(256) void fused_gemm_ln_kernel(
    const float* __restrict__ coarse_features, // [NC, CC]
    const float* __restrict__ fine_features,   // [Nf, CFD]
    const float* __restrict__ W,               // [CO, CIN] row-major
    const float* __restrict__ bias,            // [CO]
    const float* __restrict__ gamma,           // [CO]
    const float* __restrict__ beta,            // [CO]
    const int*   __restrict__ knn_idx,         // [Nf,3]
    const float* __restrict__ knn_w,           // [Nf,3]
    float* __restrict__ out,                   // [Nf, CO]
    int Nf)
{
    constexpr int XS_STRIDE = CIN + 4;  // 772 floats: rows land on distinct banks
    constexpr int HS_STRIDE = CO + 4;   // 260 floats

    __shared__ float smem[16 * XS_STRIDE];  // x-tile, later reused as h-tile
    __shared__ float ps1[256], ps2[256];
    __shared__ float smu[16], srs[16];

    const int tid = threadIdx.x;
    const int m0  = blockIdx.x * 16;

    // ---- Phase 1: assemble x-tile [16][768] = [interp(512) | fine(256)] ----
    {
        const int r  = tid >> 4;   // tile row 0..15
        const int cg = tid & 15;   // column lane 0..15
        const int n  = m0 + r;
        const int   j0 = knn_idx[n * 3 + 0];
        const int   j1 = knn_idx[n * 3 + 1];
        const int   j2 = knn_idx[n * 3 + 2];
        const float w0 = knn_w[n * 3 + 0];
        const float w1 = knn_w[n * 3 + 1];
        const float w2 = knn_w[n * 3 + 2];
        const float* f0 = coarse_features + (long)j0 * CC;
        const float* f1 = coarse_features + (long)j1 * CC;
        const float* f2 = coarse_features + (long)j2 * CC;
        float* xr = smem + r * XS_STRIDE;

        #pragma unroll 4
        for (int i = 0; i < 32; ++i) {          // interp: 512 cols / 16 threads
            const int c = cg + i * 16;          // adjacent lanes -> adjacent addrs
            xr[c] = w0 * f0[c] + w1 * f1[c] + w2 * f2[c];
        }
        const float* ff = fine_features + (long)n * CFD;
        #pragma unroll 4
        for (int i = 0; i < 16; ++i) {          // concat fine features
            const int c = cg + i * 16;
            xr[CC + c] = ff[c];
        }
    }
    __syncthreads();

    // ---- Phase 2: GEMM 16x256 += x(16x768) @ W^T via V_WMMA_F32_16X16X4_F32 ----
    const int wave = tid >> 5;      // 0..7
    const int lane = tid & 31;
    const int half = lane >> 4;     // 0 -> K{0,1}, 1 -> K{2,3}  (ISA f32 A/B layout)
    const int lr   = lane & 15;     // A: M index; B: N index within tile
    const int n0   = wave * 32;

    v8f acc0 = {0.f, 0.f, 0.f, 0.f, 0.f, 0.f, 0.f, 0.f};
    v8f acc1 = acc0;

    const float* xA  = smem + lr * XS_STRIDE + half * 2;            // A fragment base
    const float* wB0 = W + (long)(n0 + lr) * CIN + half * 2;        // B tile 0
    const float* wB1 = W + (long)(n0 + 16 + lr) * CIN + half * 2;   // B tile 1

    #pragma unroll 4
    for (int k0 = 0; k0 < CIN; k0 += 4) {
        const v2f a  = *(const v2f*)(xA  + k0);   // 8B LDS load, bank-conflict-free
        const v2f b0 = *(const v2f*)(wB0 + k0);   // 8B global load, L2-hot W
        const v2f b1 = *(const v2f*)(wB1 + k0);
        acc0 = __builtin_amdgcn_wmma_f32_16x16x4_f32(
                   false, a, false, b0, (short)0, acc0, false, false);
        acc1 = __builtin_amdgcn_wmma_f32_16x16x4_f32(
                   false, a, false, b1, (short)0, acc1, false, false);
    }

    __syncthreads();   // everyone done reading x-tile -> safe to overwrite

    // ---- Phase 3: bias + ReLU, deposit h-tile into reused LDS ----
    {
        float* hs = smem;
        const float b0v = bias[n0 + lr];
        const float b1v = bias[n0 + 16 + lr];
        #pragma unroll
        for (int v = 0; v < 8; ++v) {
            const int M = v + half * 8;          // C/D layout: VGPR v, lane half
            hs[M * HS_STRIDE + n0 + lr]      = fmaxf(acc0[v] + b0v, 0.f);
            hs[M * HS_STRIDE + n0 + 16 + lr] = fmaxf(acc1[v] + b1v, 0.f);
        }
    }
    __syncthreads();

    // ---- Phase 4: LayerNorm over 256 cols per row (two-stage reduction) ----
    {
        const int r = tid >> 4, cg = tid & 15;
        const float* hr = smem + r * HS_STRIDE + cg * 16;
        float s = 0.f, s2 = 0.f;
        #pragma unroll
        for (int j = 0; j < 16; ++j) { const float v = hr[j]; s += v; s2 += v * v; }
        ps1[tid] = s; ps2[tid] = s2;
    }
    __syncthreads();
    if (tid < 16) {
        float s = 0.f, s2 = 0.f;
        #pragma unroll
        for (int j = 0; j < 16; ++j) { s += ps1[tid * 16 + j]; s2 += ps2[tid * 16 + j]; }
        const float mu  = s * (1.0f / CO);
        const float var = fmaxf(s2 * (1.0f / CO) - mu * mu, 0.f);
        smu[tid] = mu;
        srs[tid] = rsqrtf(var + LN_EPS);
    }
    __syncthreads();
    {
        const int r = tid >> 4, cg = tid & 15;
        const float mu = smu[r], rs = srs[r];
        const float* hr = smem + r * HS_STRIDE;
        float* orow = out + (long)(m0 + r) * CO;
        #pragma unroll
        for (int j = 0; j < 16; ++j) {
            const int c = cg * 16 + j;           // 16 threads/row, 16 consecutive cols per thread
            orow[c] = gamma[c] * (hr[c] - mu) * rs + beta[c];
        }
    }
}

// ---------------------------------------------------------------------------
extern "C" void kernel_launch(void* const* d_in, const int* in_sizes, int n_in,
                              void* d_out, int out_size, void* d_ws, size_t ws_size,
                              hipStream_t stream)
{
    const float* coarse_features = (const float*)d_in[0];
    const float* coarse_pos      = (const float*)d_in[1];
    const float* fine_features   = (const float*)d_in[2];
    const float* fine_pos        = (const float*)d_in[3];
    const float* W               = (const float*)d_in[4];
    const float* b               = (const float*)d_in[5];
    const float* gamma           = (const float*)d_in[6];
    const float* beta            = (const float*)d_in[7];
    float* out = (float*)d_out;

    const int Nf = in_sizes[3] / 3;  // 16384

    int*   knn_idx = (int*)d_ws;
    float* knn_w   = (float*)((char*)d_ws + (size_t)Nf * 3 * sizeof(int));

    knn_kernel<<<(Nf + 255) / 256, 256, 0, stream>>>(
        coarse_pos, fine_pos, knn_idx, knn_w, Nf);

    fused_gemm_ln_kernel<<<Nf / 16, 256, 0, stream>>>(
        coarse_features, fine_features, W, b, gamma, beta,
        knn_idx, knn_w, out, Nf);
}